// FAMoELayer_58763742544734
// MI455X (gfx1250) — compile-verified
//
#include <hip/hip_runtime.h>
#include <hip/hip_bf16.h>
#include <math.h>

typedef __attribute__((ext_vector_type(2))) float v2f;
typedef __attribute__((ext_vector_type(8))) float v8f;

#define SEQ    50
#define FREQ   26
#define HID    1024
#define NBATCH 2048
#define KPAD   52      // K (=SEQ) padded to a multiple of 4
#define XST    52      // LDS row stride for the x tile (dwords)
#define BST    54      // LDS row stride for the DFT basis (dwords)
#define TWOPI  6.28318530717958647692f

// Dynamic LDS layout (floats):
//   sX    [HID*XST]   x[b] tile, K-padded            (212992 B)
//   sBT   [64*BST]    cos|sin basis, [n][k]          ( 13824 B)
//   sM    [64*XST]    circulant filter matrix [l][m] ( 13312 B)
//   sFreq [32]        magnitude-spectrum accumulator
//   gating scratch + irfft taps + band indices
#define SX_F    (HID * XST)
#define SBT_F   (64 * BST)
#define SM_F    (64 * XST)
#define SMEM_FLOATS (SX_F + SBT_F + SM_F + 32 + 26 + 26 + 8 + 8 + 26 + 64 + 16)
#define SMEM_BYTES  (SMEM_FLOATS * 4)

// ---------------------------------------------------------------------------
// Fused FAMoE kernel: one 32-wave workgroup per batch. x[b] (200 KB) lives in
// CDNA5's 320KB workgroup LDS for the whole kernel -> x is read from HBM once.
// ---------------------------------------------------------------------------
__global__ void __launch_bounds__(1024)
famoe_fused(const float* __restrict__ x,
            const float* __restrict__ bb,
            const float* __restrict__ W1,
            const float* __restrict__ bias1,
            const float* __restrict__ W2,
            const float* __restrict__ bias2,
            float* __restrict__ out) {
  extern __shared__ __align__(16) float smem[];
  float* sX    = smem;                 // [HID][XST]
  float* sBT   = sX + SX_F;            // [64][BST]
  float* sM    = sBT + SBT_F;          // [64][XST]
  float* sFreq = sM + SM_F;            // [32]
  float* sg    = sFreq + 32;           // [26] gating input
  float* shh   = sg + 26;              // [26] hidden
  float* slog  = shh + 26;             // [8]  logits
  float* sgate = slog + 8;             // [8]  softmax gates
  float* sw    = sgate + 8;            // [26] per-frequency weight
  float* sC    = sw + 26;              // [64] irfft taps
  int*   sidx  = (int*)(sC + 64);      // [16] band boundary bins

  const int t = threadIdx.x;           // 0..1023
  const int b = blockIdx.x;

  // ---- Phase 0: stage x[b] + DFT basis into LDS (coalesced) ---------------
  const float* xb = x + (size_t)b * HID * SEQ;
  for (int i = t; i < HID * SEQ; i += 1024) {    // 50 fully-coalesced iters
    int r = i / SEQ, c = i - r * SEQ;
    sX[r * XST + c] = xb[i];
  }
  for (int i = t; i < HID * 2; i += 1024) {      // zero K-padding cols 50..51
    int r = i >> 1;
    sX[r * XST + SEQ + (i & 1)] = 0.0f;
  }
  // basis: n<32 -> cos(2*pi*f*k/50) (Re), n>=32 -> sin (Im; sign irrelevant
  // for magnitude). Zero-padded for f>=26 and k>=50.
  for (int i = t; i < 64 * KPAD; i += 1024) {
    int n = i / KPAD, k = i - n * KPAD;
    float v = 0.0f;
    int f = n & 31;
    if (f < FREQ && k < SEQ) {
      float ang = TWOPI * (float)(f * k) / (float)SEQ;
      v = (n < 32) ? __cosf(ang) : __sinf(ang);
    }
    sBT[n * BST + k] = v;
  }
  if (t < 32) sFreq[t] = 0.0f;
  __syncthreads();

  const int w = t >> 5, lane = t & 31, half = lane >> 4, lm = lane & 15;

  // ---- Phase 1: magnitude spectrum, DFT-as-GEMM (WMMA f32 16x16x4) --------
  // Each wave: 2 tiles of 16 h-rows; N = 64 basis cols (Re 0..31 | Im 0..31).
  for (int s = 0; s < 2; ++s) {
    const int row = (w * 2 + s) * 16 + lm;       // A-fragment h-row
    v8f a0 = {}, a1 = {}, a2 = {}, a3 = {};
    #pragma unroll
    for (int kk = 0; kk < KPAD; kk += 4) {
      int k0 = kk + 2 * half;
      v2f av = *reinterpret_cast<const v2f*>(&sX[row * XST + k0]);
      v2f q0 = *reinterpret_cast<const v2f*>(&sBT[(lm)      * BST + k0]);
      v2f q1 = *reinterpret_cast<const v2f*>(&sBT[(16 + lm) * BST + k0]);
      v2f q2 = *reinterpret_cast<const v2f*>(&sBT[(32 + lm) * BST + k0]);
      v2f q3 = *reinterpret_cast<const v2f*>(&sBT[(48 + lm) * BST + k0]);
      a0 = __builtin_amdgcn_wmma_f32_16x16x4_f32(false, av, false, q0, (short)0, a0, false, false);
      a1 = __builtin_amdgcn_wmma_f32_16x16x4_f32(false, av, false, q1, (short)0, a1, false, false);
      a2 = __builtin_amdgcn_wmma_f32_16x16x4_f32(false, av, false, q2, (short)0, a2, false, false);
      a3 = __builtin_amdgcn_wmma_f32_16x16x4_f32(false, av, false, q3, (short)0, a3, false, false);
    }
    float m0 = 0.0f, m1 = 0.0f;                  // partial sums over 8 rows
    #pragma unroll
    for (int i = 0; i < 8; ++i) {
      m0 += sqrtf(a0[i] * a0[i] + a2[i] * a2[i]);   // freq = lm
      m1 += sqrtf(a1[i] * a1[i] + a3[i] * a3[i]);   // freq = 16+lm (0 if >=26)
    }
    atomicAdd(&sFreq[lm], m0);                   // ds_add_f32
    atomicAdd(&sFreq[16 + lm], m1);
  }
  __syncthreads();

  // ---- Phase 2: gating MLP -> softmax -> band weights -> irfft taps -------
  if (t < FREQ) sg[t] = sFreq[t] * (1.0f / (float)HID);
  if (t == 0) {
    float s[7];
    for (int i = 0; i < 7; ++i) s[i] = 1.0f / (1.0f + __expf(-bb[i]));
    for (int i = 1; i < 7; ++i) {                // insertion sort (ascending)
      float key = s[i]; int j = i - 1;
      while (j >= 0 && s[j] > key) { s[j + 1] = s[j]; --j; }
      s[j + 1] = key;
    }
    sidx[0] = 0;
    for (int i = 0; i < 7; ++i) sidx[i + 1] = (int)(s[i] * (float)FREQ);
    sidx[8] = FREQ;
  }
  __syncthreads();
  if (t < FREQ) {                                // h = relu(gi @ W1 + b1)
    float hv = bias1[t];
    for (int i = 0; i < FREQ; ++i) hv += sg[i] * W1[i * FREQ + t];
    shh[t] = fmaxf(hv, 0.0f);
  }
  __syncthreads();
  if (t < 8) {                                   // logits = h @ W2 + b2
    float lg = bias2[t];
    for (int j = 0; j < FREQ; ++j) lg += shh[j] * W2[j * 8 + t];
    slog[t] = lg;
  }
  __syncthreads();
  if (t == 0) {                                  // softmax over 8 experts
    float mx = slog[0];
    for (int e = 1; e < 8; ++e) mx = fmaxf(mx, slog[e]);
    float ex[8], ssum = 0.0f;
    for (int e = 0; e < 8; ++e) { ex[e] = __expf(slog[e] - mx); ssum += ex[e]; }
    float inv = 1.0f / ssum;
    for (int e = 0; e < 8; ++e) sgate[e] = ex[e] * inv;
  }
  __syncthreads();
  if (t < FREQ) {                                // w[k] = sum_e gate[e]*mask
    float wv = 0.0f;
    for (int e = 0; e < 8; ++e)
      if (t >= sidx[e] && t < sidx[e + 1]) wv += sgate[e];
    sw[t] = wv;
  }
  __syncthreads();
  if (t < 64) {                                  // c = irfft(w): 50 real taps
    float cv = 0.0f;
    if (t < SEQ) {
      cv = sw[0];
      for (int k = 1; k < 25; ++k)
        cv += 2.0f * sw[k] * __cosf(TWOPI * (float)(k * t) / (float)SEQ);
      cv += sw[25] * ((t & 1) ? -1.0f : 1.0f);   // Nyquist bin, counted once
      cv *= (1.0f / (float)SEQ);
    }
    sC[t] = cv;
  }
  __syncthreads();
  for (int i = t; i < 64 * KPAD; i += 1024) {    // circulant M[l][m]=c[(l-m)%50]
    int l = i / KPAD, m = i - l * KPAD;
    float v = 0.0f;
    if (l < SEQ && m < SEQ) {
      int d = l - m; if (d < 0) d += SEQ;
      v = sC[d];
    }
    sM[l * XST + m] = v;
  }
  __syncthreads();

  // ---- Phase 3: Out[b] (50x1024) = M[b] @ X[b]^T, x re-read from LDS ------
  float* ob = out + (size_t)b * SEQ * HID;
  for (int s = 0; s < 2; ++s) {
    const int hcol = (w * 2 + s) * 16 + lm;      // B-fragment h-column
    v8f acc[4] = {v8f{}, v8f{}, v8f{}, v8f{}};   // 4 l-tiles x 16 h-cols
    #pragma unroll
    for (int kk = 0; kk < KPAD; kk += 4) {
      int k0 = kk + 2 * half;
      v2f bv = *reinterpret_cast<const v2f*>(&sX[hcol * XST + k0]);
      #pragma unroll
      for (int mt = 0; mt < 4; ++mt) {
        v2f av = *reinterpret_cast<const v2f*>(&sM[(mt * 16 + lm) * XST + k0]);
        acc[mt] = __builtin_amdgcn_wmma_f32_16x16x4_f32(false, av, false, bv,
                                                        (short)0, acc[mt], false, false);
      }
    }
    #pragma unroll
    for (int mt = 0; mt < 4; ++mt) {
      #pragma unroll
      for (int i = 0; i < 8; ++i) {
        int l = mt * 16 + half * 8 + i;          // C/D layout: M = vgpr + 8*(lane/16)
        if (l < SEQ)
          ob[l * HID + hcol] = acc[mt][i];
      }
    }
  }
}

// ---------------------------------------------------------------------------
extern "C" void kernel_launch(void* const* d_in, const int* in_sizes, int n_in,
                              void* d_out, int out_size, void* d_ws, size_t ws_size,
                              hipStream_t stream) {
  const float* x     = (const float*)d_in[0];
  const float* bb    = (const float*)d_in[1];
  const float* W1    = (const float*)d_in[2];
  const float* bias1 = (const float*)d_in[3];
  const float* W2    = (const float*)d_in[4];
  const float* bias2 = (const float*)d_in[5];
  float* out = (float*)d_out;

  // Allow >64KB dynamic LDS (CDNA5: up to 320KB per workgroup).
  (void)hipFuncSetAttribute(reinterpret_cast<const void*>(famoe_fused),
                            hipFuncAttributeMaxDynamicSharedMemorySize,
                            SMEM_BYTES);

  famoe_fused<<<NBATCH, 1024, SMEM_BYTES, stream>>>(x, bb, W1, bias1, W2, bias2, out);
}